// MagneticFluxAttention_50122268344507
// MI455X (gfx1250) — compile-verified
//
#include <hip/hip_runtime.h>
#include <cmath>

// Problem constants (reference: x shape (4, 4096, 2) f32)
#define NB 4
#define SS 4096
#define NPAIR ((size_t)SS * (size_t)SS)
#define EPSF 1e-8f

// ws layout in u32 units:
//   [0, 2048)      : hist[NB][2][256]   (rank k and k+1 histograms)
//   [2048, 2056)   : prefix[NB][2]      (accumulated high bits of order stat)
//   [2056, 2064)   : rem[NB][2]         (remaining rank within prefix group)
//   [2064, 2068)   : thr[NB]            (float threshold per batch)
#define WS_HIST 0
#define WS_PFX  2048
#define WS_REM  2056
#define WS_THR  2064

typedef float v2f __attribute__((ext_vector_type(2)));
typedef float v8f __attribute__((ext_vector_type(8)));

#if defined(__has_builtin)
#if __has_builtin(__builtin_amdgcn_wmma_f32_16x16x4_f32)
#define HAVE_WMMA_F32X4 1
#endif
#endif

// attention value from Gram dot + per-point radii.
// cos(phi_i - phi_j) = dot/(ri*rj); |xi-xj|^2 = ri2 + rj2 - 2 dot.
// Clamp cos_term >= 0 so all attention values are non-negative (bit-pattern
// order == numeric order for the radix select).
__device__ __forceinline__ float mf_attn(float dot, float ri2, float invri,
                                         float rj2, float invrj) {
    float cosv = fmaf(0.5f * dot, invri * invrj, 0.5f);
    cosv = fmaxf(cosv, 0.0f);
    float d2 = ri2 + rj2 - 2.0f * dot;
    d2 = fmaxf(d2, 0.0f);
    return cosv / (sqrtf(d2) + EPSF);
}

// ---------------- init: zero histograms, seed ranks k and k+1 ----------------
__global__ void mf_init(unsigned* __restrict__ ws, unsigned k) {
    int t = threadIdx.x;
    for (int i = t; i < 2048; i += 256) ws[WS_HIST + i] = 0u;
    if (t < NB * 2) {
        ws[WS_PFX + t] = 0u;
        ws[WS_REM + t] = k + (unsigned)(t & 1);  // rank k for r=0, k+1 for r=1
    }
}

// ---------------- radix-select histogram pass (8 bits per pass) --------------
__global__ void mf_hist(const float* __restrict__ x, unsigned* __restrict__ ws,
                        int p) {
    __shared__ unsigned lh[2 * 256];
    const int b = blockIdx.y;
    const int tid = threadIdx.x;
    for (int i = tid; i < 512; i += 256) lh[i] = 0u;
    __syncthreads();

    const unsigned pfx0 = ws[WS_PFX + b * 2 + 0];
    const unsigned pfx1 = ws[WS_PFX + b * 2 + 1];
    const int shift = 24 - 8 * p;
    const float* xb = x + (size_t)b * SS * 2;

    const size_t chunk = NPAIR / gridDim.x;
    const size_t start = chunk * (size_t)blockIdx.x;
    const size_t end = start + chunk;
    for (size_t idx = start + (size_t)tid; idx < end; idx += 256) {
        const int i = (int)(idx >> 12);
        const int j = (int)(idx & (size_t)(SS - 1));
        const float xi = xb[2 * i], yi = xb[2 * i + 1];
        const float xj = xb[2 * j], yj = xb[2 * j + 1];
        const float ri2 = fmaf(xi, xi, yi * yi);
        const float rj2 = fmaf(xj, xj, yj * yj);
        const float dot = fmaf(xi, xj, yi * yj);
        const float a = mf_attn(dot, ri2, rsqrtf(ri2), rj2, rsqrtf(rj2));
        const unsigned u = __float_as_uint(a);
        const unsigned bin = (u >> shift) & 255u;
        bool m0 = true, m1 = true;
        if (p != 0) {
            const unsigned hi = u >> (shift + 8);
            m0 = (hi == pfx0);
            m1 = (hi == pfx1);
        }
        if (m0) atomicAdd(&lh[bin], 1u);
        if (m1) atomicAdd(&lh[256 + bin], 1u);
    }
    __syncthreads();
    for (int i = tid; i < 512; i += 256) {
        const unsigned c = lh[i];
        if (c) atomicAdd(&ws[WS_HIST + (b * 2) * 256 + i], c);
    }
}

// -------- scan 256-bin histogram, refine prefix/rank; finalize threshold -----
__global__ void mf_select(unsigned* __restrict__ ws, int p, float frac) {
    const int b = blockIdx.x;
    if (threadIdx.x == 0) {
        for (int r = 0; r < 2; ++r) {
            const unsigned* h = &ws[WS_HIST + (b * 2 + r) * 256];
            unsigned rem = ws[WS_REM + b * 2 + r];
            unsigned pfx = ws[WS_PFX + b * 2 + r];
            unsigned cum = 0, bin = 255;
            for (int i = 0; i < 256; ++i) {
                const unsigned c = h[i];
                if (cum + c > rem) { bin = (unsigned)i; break; }
                cum += c;
            }
            ws[WS_PFX + b * 2 + r] = (pfx << 8) | bin;
            ws[WS_REM + b * 2 + r] = rem - cum;
        }
        if (p == 3) {
            const float v0 = __uint_as_float(ws[WS_PFX + b * 2 + 0]);
            const float v1 = __uint_as_float(ws[WS_PFX + b * 2 + 1]);
            ((float*)ws)[WS_THR + b] = v0 + frac * (v1 - v0);  // jnp 'linear'
        }
    }
    __syncthreads();
    for (int i = threadIdx.x; i < 512; i += blockDim.x)
        ws[WS_HIST + (b * 2) * 256 + i] = 0u;  // reset for next pass
}

// -------- fused WMMA kernel: masked row sums (phase 1) + output (phase 2) ----
// grid = (SS/16, NB), block = 256 (8 waves). Each block owns a 16-row stripe;
// each wave computes 16x16 tiles via V_WMMA_F32_16X16X4_F32 (K=2 used, 2 zero).
__global__ void mf_out(const float* __restrict__ x, float* __restrict__ out,
                       const unsigned* __restrict__ ws) {
    __shared__ float lpart[256][8];  // per-thread partial row sums
    __shared__ float rinv[16];       // 1 / (rowsum + EPS)

    const int b = blockIdx.y;
    const int i0 = blockIdx.x * 16;
    const int tid = threadIdx.x;
    const int w = tid >> 5;
    const int lane = tid & 31;
    const int half = lane >> 4;
    const int ln = lane & 15;
    const float thr = ((const float*)ws)[WS_THR + b];
    const float* xb = x + (size_t)b * SS * 2;

    // A matrix (16x4 f32): lanes 0-15 hold row ln, K=0,1; lanes 16-31 hold
    // K=2,3 which we zero (effective K=2).
    v2f A;
    A.x = (half == 0) ? xb[(size_t)(i0 + ln) * 2 + 0] : 0.0f;
    A.y = (half == 0) ? xb[(size_t)(i0 + ln) * 2 + 1] : 0.0f;

    // Per-lane row parameters: C/D row M = v + 8*half (ISA 16x16 f32 layout).
    float rxv[8], ryv[8], ri2[8], invri[8];
#pragma unroll
    for (int v = 0; v < 8; ++v) {
        const int m = v + 8 * half;
        const float xi = xb[(size_t)(i0 + m) * 2 + 0];
        const float yi = xb[(size_t)(i0 + m) * 2 + 1];
        rxv[v] = xi; ryv[v] = yi;
        ri2[v] = fmaf(xi, xi, yi * yi);
        invri[v] = rsqrtf(ri2[v]);
    }

    const int TILES = SS / 16;  // 256
    float acc[8];
#pragma unroll
    for (int v = 0; v < 8; ++v) acc[v] = 0.0f;

    // ---- phase 1: masked row sums over all column tiles ----
    for (int ct = w; ct < TILES; ct += 8) {
        const int j0 = ct * 16;
        const float xj = xb[(size_t)(j0 + ln) * 2 + 0];
        const float yj = xb[(size_t)(j0 + ln) * 2 + 1];
        const float rj2 = fmaf(xj, xj, yj * yj);
        const float invrj = rsqrtf(rj2);
        float dvec[8];
#if defined(HAVE_WMMA_F32X4)
        v2f Bm;  // B 4x16 f32: VGPR0 = K=0 (lanes0-15) / K=2 (lanes16-31), VGPR1 = K=1 / K=3
        Bm.x = (half == 0) ? xj : 0.0f;
        Bm.y = (half == 0) ? yj : 0.0f;
        v8f c = {};
        v8f d = __builtin_amdgcn_wmma_f32_16x16x4_f32(false, A, false, Bm,
                                                      (short)0, c, false, false);
#pragma unroll
        for (int v = 0; v < 8; ++v) dvec[v] = d[v];
#else
#pragma unroll
        for (int v = 0; v < 8; ++v) dvec[v] = fmaf(rxv[v], xj, ryv[v] * yj);
#endif
#pragma unroll
        for (int v = 0; v < 8; ++v) {
            const float a = mf_attn(dvec[v], ri2[v], invri[v], rj2, invrj);
            if (a > thr) acc[v] += a;
        }
    }
#pragma unroll
    for (int v = 0; v < 8; ++v) lpart[tid][v] = acc[v];
    __syncthreads();

    // Deterministic row-sum reduction: row m = v + 8*half gathers the 16 lanes
    // of the matching half in each of the 8 waves, in fixed order.
    if (tid < 16) {
        const int v = tid & 7;
        const int h = tid >> 3;
        float s = 0.0f;
        for (int ww = 0; ww < 8; ++ww)
            for (int l2 = 0; l2 < 16; ++l2)
                s += lpart[ww * 32 + h * 16 + l2][v];
        rinv[tid] = 1.0f / (s + EPSF);
    }
    __syncthreads();

    float rn[8];
#pragma unroll
    for (int v = 0; v < 8; ++v) rn[v] = rinv[v + 8 * half];

    // ---- phase 2: recompute tiles, write normalized attention + mask ----
    const size_t base = (size_t)b * SS * (size_t)SS;
    const size_t maskoff = (size_t)NB * SS * (size_t)SS;
    for (int ct = w; ct < TILES; ct += 8) {
        const int j0 = ct * 16;
        const float xj = xb[(size_t)(j0 + ln) * 2 + 0];
        const float yj = xb[(size_t)(j0 + ln) * 2 + 1];
        const float rj2 = fmaf(xj, xj, yj * yj);
        const float invrj = rsqrtf(rj2);
        float dvec[8];
#if defined(HAVE_WMMA_F32X4)
        v2f Bm;
        Bm.x = (half == 0) ? xj : 0.0f;
        Bm.y = (half == 0) ? yj : 0.0f;
        v8f c = {};
        v8f d = __builtin_amdgcn_wmma_f32_16x16x4_f32(false, A, false, Bm,
                                                      (short)0, c, false, false);
#pragma unroll
        for (int v = 0; v < 8; ++v) dvec[v] = d[v];
#else
#pragma unroll
        for (int v = 0; v < 8; ++v) dvec[v] = fmaf(rxv[v], xj, ryv[v] * yj);
#endif
#pragma unroll
        for (int v = 0; v < 8; ++v) {
            const float a = mf_attn(dvec[v], ri2[v], invri[v], rj2, invrj);
            const bool keep = (a > thr);
            const float mk = keep ? 1.0f : 0.0f;
            const float av = (keep ? a : 0.0f) * rn[v];
            const size_t idx =
                base + (size_t)(i0 + v + 8 * half) * SS + (size_t)(j0 + ln);
            __builtin_nontemporal_store(av, &out[idx]);
            __builtin_nontemporal_store(mk, &out[maskoff + idx]);
        }
    }
}

extern "C" void kernel_launch(void* const* d_in, const int* in_sizes, int n_in,
                              void* d_out, int out_size, void* d_ws,
                              size_t ws_size, hipStream_t stream) {
    (void)in_sizes; (void)n_in; (void)out_size; (void)ws_size;
    const float* x = (const float*)d_in[0];
    float* out = (float*)d_out;
    unsigned* ws = (unsigned*)d_ws;

    // quantile position: q = 1 - SPARSITY, idx = q*(N-1), linear interpolation
    const double PHI = (1.0 + std::sqrt(5.0)) / 2.0;
    const double PI = 3.14159265358979323846;
    const double E = 2.71828182845904523536;
    const double SPARSITY = 4.0 * PHI / (3.0 * PI * std::sqrt(E));
    const double q = 1.0 - SPARSITY;
    const double pos = q * (double)(NPAIR - 1);
    const double kf = std::floor(pos);
    const unsigned k = (unsigned)kf;
    const float frac = (float)(pos - kf);

    mf_init<<<1, 256, 0, stream>>>(ws, k);
    for (int p = 0; p < 4; ++p) {
        mf_hist<<<dim3(64, NB), 256, 0, stream>>>(x, ws, p);
        mf_select<<<dim3(NB), 64, 0, stream>>>(ws, p, frac);
    }
    mf_out<<<dim3(SS / 16, NB), 256, 0, stream>>>(x, out, ws);
}